// ExpectationSoftmaxLayer_17746804867877
// MI455X (gfx1250) — compile-verified
//
#include <hip/hip_runtime.h>
#include <cstdint>
#include <cstddef>

// Problem constants (from reference)
#define BATCH   256
#define IN_DIM  1024
#define OUT_DIM 1024

// Tiling
#define BT      8            // batches per block (x tile staged in LDS)
#define OPB     32           // outputs per block
#define WAVES   8
#define THREADS (WAVES * 32) // 256 threads = 8 wave32

typedef float float4v __attribute__((ext_vector_type(4)));
typedef int   int4v   __attribute__((ext_vector_type(4)));

#if __has_builtin(__builtin_amdgcn_global_load_async_to_lds_b128)
typedef __attribute__((address_space(1))) int4v as1_int4;
typedef __attribute__((address_space(3))) int4v as3_int4;
#define HAVE_ASYNC_LDS 1
#endif

__device__ __forceinline__ float fast_exp2(float v) {
#if __has_builtin(__builtin_amdgcn_exp2f)
    return __builtin_amdgcn_exp2f(v);   // v_exp_f32
#else
    return exp2f(v);
#endif
}

__global__ __launch_bounds__(THREADS)
void expectation_softmax_kernel(const float* __restrict__ x,
                                const float* __restrict__ weight,
                                const float* __restrict__ log_tau,
                                float* __restrict__ out)
{
    __shared__ float xs[BT * IN_DIM];   // 32 KB x tile (LDS is 320KB/WGP)

    const int bt   = blockIdx.x % (BATCH / BT);
    const int og   = blockIdx.x / (BATCH / BT);
    const int b0   = bt * BT;
    const int o0   = og * OPB;
    const int lane = threadIdx.x & 31;
    const int wave = threadIdx.x >> 5;

    // tau folded into base-2 softmax: exp(tau*t) == 2^(tau*log2e*t)
    const float c = expf(log_tau[0]) * 1.44269504088896340736f;

    // ---- Stage x[b0:b0+BT, :] into LDS -------------------------------------
    const float* gx = x + (size_t)b0 * IN_DIM;
    {
#ifdef HAVE_ASYNC_LDS
        // CDNA5 async global->LDS (ASYNCcnt) copy: 2048 float4 / 256 threads
        #pragma unroll
        for (int t = 0; t < (BT * IN_DIM / 4) / THREADS; ++t) {
            const int idx4 = t * THREADS + threadIdx.x;
            as1_int4* g = (as1_int4*)(uintptr_t)(const void*)(gx + idx4 * 4);
            // low 32 bits of a generic LDS pointer are the ds-relative offset
            as3_int4* l = (as3_int4*)(uint32_t)(uintptr_t)(void*)(&xs[idx4 * 4]);
            __builtin_amdgcn_global_load_async_to_lds_b128(g, l, 0, 0);
        }
#if __has_builtin(__builtin_amdgcn_s_wait_asynccnt)
        __builtin_amdgcn_s_wait_asynccnt(0);
#else
        asm volatile("s_wait_asynccnt 0" ::: "memory");
#endif
#else
        #pragma unroll
        for (int t = 0; t < (BT * IN_DIM / 4) / THREADS; ++t) {
            const int idx4 = t * THREADS + threadIdx.x;
            ((float4v*)xs)[idx4] = ((const float4v*)gx)[idx4];
        }
#endif
    }
    __syncthreads();

    // ---- Main loop: each wave owns OPB/WAVES outputs for this batch tile ---
    for (int oo = wave; oo < OPB; oo += WAVES) {
        const int o = o0 + oo;
        const float* wrow = weight + (size_t)o * IN_DIM;

        // Prefetch the next weight row this wave will touch (global_prefetch_b8)
        if (oo + WAVES < OPB)
            __builtin_prefetch(weight + (size_t)(o + WAVES) * IN_DIM + lane * 32, 0, 1);

        // Load + leaky_clamp the weight row once; reuse across BT batches.
        // lane reads float4 index (j*32 + lane): fully coalesced b128 loads.
        float4v aw[8];
        #pragma unroll
        for (int j = 0; j < 8; ++j) {
            float4v w = ((const float4v*)wrow)[j * 32 + lane];
            #pragma unroll
            for (int k = 0; k < 4; ++k) {
                const float v = w[k];
                float r = v;
                r = (v < 0.0f) ? 0.1f * v : r;          // lo + slope*(v-lo), lo=0
                r = (v > 1.0f) ? 0.9f + 0.1f * v : r;   // hi + slope*(v-hi), hi=1
                w[k] = r;
            }
            aw[j] = w;
        }

        #pragma unroll 1
        for (int bb = 0; bb < BT; ++bb) {
            const float4v* xsrow = (const float4v*)(xs + bb * IN_DIM);

            // z = x * aw, kept in registers; per-component running max (4 chains)
            float4v z[8];
            float4v mv = {-3.402823466e38f, -3.402823466e38f,
                          -3.402823466e38f, -3.402823466e38f};
            #pragma unroll
            for (int j = 0; j < 8; ++j) {
                const float4v xv = xsrow[j * 32 + lane];
                const float4v zv = aw[j] * xv;
                z[j] = zv;
                mv[0] = fmaxf(mv[0], zv[0]);
                mv[1] = fmaxf(mv[1], zv[1]);
                mv[2] = fmaxf(mv[2], zv[2]);
                mv[3] = fmaxf(mv[3], zv[3]);
            }
            float m = fmaxf(fmaxf(mv[0], mv[1]), fmaxf(mv[2], mv[3]));
            // wave32 butterfly max
            #pragma unroll
            for (int s = 16; s > 0; s >>= 1)
                m = fmaxf(m, __shfl_xor(m, s, 32));

            // denominator and numerator; 4 independent accumulator chains each
            float4v dv = {0.0f, 0.0f, 0.0f, 0.0f};
            float4v nv = {0.0f, 0.0f, 0.0f, 0.0f};
            #pragma unroll
            for (int j = 0; j < 8; ++j) {
                #pragma unroll
                for (int k = 0; k < 4; ++k) {
                    const float zz = z[j][k];
                    const float e  = fast_exp2(c * (zz - m));  // v_exp_f32
                    dv[k] += e;
                    nv[k] = fmaf(e, zz, nv[k]);
                }
            }
            float d = (dv[0] + dv[1]) + (dv[2] + dv[3]);
            float n = (nv[0] + nv[1]) + (nv[2] + nv[3]);
            #pragma unroll
            for (int s = 16; s > 0; s >>= 1) {
                d += __shfl_xor(d, s, 32);
                n += __shfl_xor(n, s, 32);
            }

            if (lane == 0)
                out[(size_t)(b0 + bb) * OUT_DIM + o] = n / d;
        }
    }
}

extern "C" void kernel_launch(void* const* d_in, const int* in_sizes, int n_in,
                              void* d_out, int out_size, void* d_ws, size_t ws_size,
                              hipStream_t stream) {
    const float* x       = (const float*)d_in[0];  // (B, IN)  f32
    const float* weight  = (const float*)d_in[1];  // (OUT, IN) f32
    const float* log_tau = (const float*)d_in[2];  // scalar   f32
    float* out           = (float*)d_out;          // (B, OUT) f32

    const int grid = (OUT_DIM / OPB) * (BATCH / BT);  // 32 * 32 = 1024 blocks
    expectation_softmax_kernel<<<grid, THREADS, 0, stream>>>(x, weight, log_tau, out);
}